// EdgeConvLayer_85435489452613
// MI455X (gfx1250) — compile-verified
//
#include <hip/hip_runtime.h>
#include <hip/hip_bf16.h>

// EdgeConv on MI455X (gfx1250, wave32) using V_WMMA_F32_16X16X4_F32.
//
// Per 16-edge tile (one wave):
//   A (16x144) = [x[src] | x[dst] | edge_attr], gathered directly from global
//   H = relu(A @ W1 + b1)          -> 36 k-steps x 4 N-tiles WMMA
//   M = H @ W2 + b2                -> 16 k-steps x 4 N-tiles WMMA
//   out[dst] += M  (coalesced fp32 global atomics)
//
// All 36 A-fragment gathers are hoisted ahead of the WMMA chain so the
// scheduler can keep 36 global_load_b64 in flight (in-order LOADcnt waits
// overlap the matrix pipe). Both GEMM k-loops fully unrolled.
//
// WMMA f32 16x16x4 register layouts (ISA 7.12.2):
//   A: lane L holds row m=L&15, elems K = 4k + 2*(L>>4) + {0,1}  (contiguous)
//   B: lane L holds col n=L&15, elems K = 4k + 2*(L>>4) + {0,1}
//   C/D: vgpr r, lane L -> row r + 8*(L>>4), col L&15

typedef float v2f __attribute__((ext_vector_type(2)));
typedef float v8f __attribute__((ext_vector_type(8)));

#define IN_F   64
#define EDGE_F 16
#define D_IN   144          // 2*IN_F + EDGE_F
#define OUT_F  64
#define K1     36           // D_IN / 4
#define K2     16           // OUT_F / 4
#define HS     66           // padded H row stride (floats, even for b64 loads)
#define WAVES  2            // waves per block (keeps static LDS < 64 KB)

__global__ __launch_bounds__(32 * WAVES)
void edgeconv_wmma_f32_kernel(const float* __restrict__ x,
                              const int*   __restrict__ eidx,   // (2,E) flat
                              const float* __restrict__ eattr,
                              const float* __restrict__ W1,
                              const float* __restrict__ b1,
                              const float* __restrict__ W2,
                              const float* __restrict__ b2,
                              float*       __restrict__ out,
                              int E, int nTiles)
{
    // Weights in K-pair-interleaved layout: sW[(p*64 + n)*2 + j] = W[(2p+j)*64 + n]
    __shared__ float sW1[D_IN * OUT_F];      // 36864 B
    __shared__ float sW2[OUT_F * OUT_F];     // 16384 B
    __shared__ float sB1[OUT_F];
    __shared__ float sB2[OUT_F];
    __shared__ float sH[WAVES][16 * HS];     // 8448 B: GEMM1->GEMM2 transpose
    __shared__ int   sSrc[WAVES][16];
    __shared__ int   sDst[WAVES][16];

    const int tid  = threadIdx.x;
    const int lane = tid & 31;
    const int wid  = tid >> 5;
    const int m    = lane & 15;   // row (A) / col (B,C,D) index
    const int hi   = lane >> 4;   // K-pair / row-half selector

    // ---- cooperative weight load into interleaved LDS layout ----
    for (int i = tid; i < D_IN * OUT_F; i += 32 * WAVES) {
        int p = i >> 7;           // K-pair index
        int n = (i >> 1) & 63;    // column
        int j = i & 1;            // row within pair
        sW1[i] = W1[(2 * p + j) * OUT_F + n];
    }
    for (int i = tid; i < OUT_F * OUT_F; i += 32 * WAVES) {
        int p = i >> 7;
        int n = (i >> 1) & 63;
        int j = i & 1;
        sW2[i] = W2[(2 * p + j) * OUT_F + n];
    }
    for (int i = tid; i < OUT_F; i += 32 * WAVES) {
        sB1[i] = b1[i];
        sB2[i] = b2[i];
    }
    __syncthreads();

    const v8f vzero = {0.f, 0.f, 0.f, 0.f, 0.f, 0.f, 0.f, 0.f};

    for (int tile = blockIdx.x * WAVES + wid; tile < nTiles;
         tile += gridDim.x * WAVES) {
        const int e0 = tile << 4;

        // ---- edge indices for the 16 rows of this tile ----
        if (lane < 16) {
            int e = e0 + lane;
            int s = 0, d = 0;
            if (e < E) { s = eidx[e]; d = eidx[E + e]; }
            sSrc[wid][lane] = s;
            sDst[wid][lane] = d;
        }
        __builtin_amdgcn_wave_barrier();   // wave-synchronous LDS, keep order

        const int srow = sSrc[wid][m];
        const int drow = sDst[wid][m];
        const int erow = (e0 + m < E) ? (e0 + m) : 0;

        // per-lane A-row base pointers (already offset by 2*hi -> b64 loads)
        const float* aS = x + (size_t)srow * IN_F + 2 * hi;
        const float* aD = x + (size_t)drow * IN_F + 2 * hi;
        const float* aE = eattr + (size_t)erow * EDGE_F + 2 * hi;

        // ---- hoisted A-fragment gathers: 36 independent b64 loads in flight ----
        v2f afr[K1];
        #pragma unroll
        for (int k = 0; k < 16; ++k) afr[k]      = *(const v2f*)(aS + (k << 2));
        #pragma unroll
        for (int k = 0; k < 16; ++k) afr[16 + k] = *(const v2f*)(aD + (k << 2));
        #pragma unroll
        for (int k = 0; k < 4;  ++k) afr[32 + k] = *(const v2f*)(aE + (k << 2));

        // ---- GEMM1: H = A(16x144) @ W1(144x64) ----
        v8f acc[4];
        #pragma unroll
        for (int t = 0; t < 4; ++t) acc[t] = vzero;

        #pragma unroll
        for (int k = 0; k < K1; ++k) {
            const int pb = ((k << 1) + hi) << 7;   // (p*64)*2
            #pragma unroll
            for (int t = 0; t < 4; ++t) {
                v2f b = *(const v2f*)&sW1[pb + (((t << 4) + m) << 1)];
                acc[t] = __builtin_amdgcn_wmma_f32_16x16x4_f32(
                    false, afr[k], false, b, (short)0, acc[t], false, false);
            }
        }

        // ---- bias + ReLU, spill H to LDS (D-layout -> row-major) ----
        float* hBase = &sH[wid][0];
        #pragma unroll
        for (int t = 0; t < 4; ++t) {
            const float bb = sB1[(t << 4) + m];
            #pragma unroll
            for (int r = 0; r < 8; ++r) {
                float v = acc[t][r] + bb;
                v = v > 0.f ? v : 0.f;
                hBase[(r + (hi << 3)) * HS + (t << 4) + m] = v;
            }
        }
        __builtin_amdgcn_wave_barrier();

        // ---- GEMM2: M = H(16x64) @ W2(64x64) ----
        v8f acc2[4];
        #pragma unroll
        for (int t = 0; t < 4; ++t) acc2[t] = vzero;

        const float* hRow = &sH[wid][m * HS + 2 * hi];
        #pragma unroll
        for (int k = 0; k < K2; ++k) {
            v2f a = *(const v2f*)(hRow + (k << 2));
            const int pb = ((k << 1) + hi) << 7;
            #pragma unroll
            for (int t = 0; t < 4; ++t) {
                v2f b = *(const v2f*)&sW2[pb + (((t << 4) + m) << 1)];
                acc2[t] = __builtin_amdgcn_wmma_f32_16x16x4_f32(
                    false, a, false, b, (short)0, acc2[t], false, false);
            }
        }

        // ---- bias + scatter-add to out[dst] ----
        // For each r, lanes 0-15 share one edge -> 16 consecutive floats of
        // out[dst]: coalesced global_atomic_add_f32.
        #pragma unroll
        for (int r = 0; r < 8; ++r) {
            const int row = r + (hi << 3);
            const int e   = e0 + row;
            if (e < E) {
                const int dn = sDst[wid][row];
                float* op = out + (size_t)dn * OUT_F + m;
                #pragma unroll
                for (int t = 0; t < 4; ++t) {
                    atomicAdd(op + (t << 4), acc2[t][r] + sB2[(t << 4) + m]);
                }
            }
        }
    }
}

extern "C" void kernel_launch(void* const* d_in, const int* in_sizes, int n_in,
                              void* d_out, int out_size, void* d_ws, size_t ws_size,
                              hipStream_t stream) {
    const float* x     = (const float*)d_in[0];
    const int*   eidx  = (const int*)  d_in[1];   // (2,E), int32 (JAX demotes i64)
    const float* eattr = (const float*)d_in[2];
    const float* W1    = (const float*)d_in[3];
    const float* b1    = (const float*)d_in[4];
    const float* W2    = (const float*)d_in[5];
    const float* b2    = (const float*)d_in[6];
    float* out = (float*)d_out;

    const int E      = in_sizes[1] / 2;
    const int nTiles = (E + 15) / 16;

    // segment_sum semantics: nodes with no incoming edges must be zero.
    hipMemsetAsync(d_out, 0, (size_t)out_size * sizeof(float), stream);

    int blocks = 2560;                            // persistent: ~10 tiles/wave
    const int need = (nTiles + WAVES - 1) / WAVES;
    if (blocks > need) blocks = need;
    if (blocks < 1) blocks = 1;

    edgeconv_wmma_f32_kernel<<<blocks, 32 * WAVES, 0, stream>>>(
        x, eidx, eattr, W1, b1, W2, b2, out, E, nTiles);
}